// GIN_4011499454824
// MI455X (gfx1250) — compile-verified
//
#include <hip/hip_runtime.h>

#define NN 50000
#define EE 600000
#define DD 128
#define LL 3
#define GG 256
#define CC 10
#define BN_EPS 1e-5f

typedef __attribute__((ext_vector_type(2))) float v2f;
typedef __attribute__((ext_vector_type(8))) float v8f;

// ---------------- utility kernels ----------------

__global__ void k_zero(float* __restrict__ p, int n) {
  int i = blockIdx.x * blockDim.x + threadIdx.x;
  if (i < n) p[i] = 0.f;
}

// scatter-add: out[dst[e]] += x[src[e]] * (w ? w[e] : 1).  32 threads per edge,
// each handles a float4 chunk (coalesced), 4 global_atomic_add_f32 each.
__global__ void k_scatter(const float* __restrict__ x,
                          const int* __restrict__ src,
                          const int* __restrict__ dst,
                          const float* __restrict__ w,
                          float* __restrict__ out) {
  int t = blockIdx.x * blockDim.x + threadIdx.x;
  int e = t >> 5, q = t & 31;
  if (e >= EE) return;
  int s = src[e], d = dst[e];
  float wt = w ? w[e] : 1.0f;
  float4 v = ((const float4*)(x + (size_t)s * DD))[q];
  float* o = out + (size_t)d * DD + (size_t)q * 4;
  atomicAdd(o + 0, v.x * wt);
  atomicAdd(o + 1, v.y * wt);
  atomicAdd(o + 2, v.z * wt);
  atomicAdd(o + 3, v.w * wt);
}

// out[n] = mask[n] ? x2[n] : x[n]   (per-row select, float4 chunks)
__global__ void k_select(const float* __restrict__ x,
                         const float* __restrict__ x2,
                         const int* __restrict__ mask,
                         float* __restrict__ out) {
  int t = blockIdx.x * blockDim.x + threadIdx.x;
  int node = t >> 5, q = t & 31;
  if (node >= NN) return;
  const float* srcp = mask[node] ? x2 : x;
  float4 v = ((const float4*)(srcp + (size_t)node * DD))[q];
  ((float4*)(out + (size_t)node * DD))[q] = v;
}

// global add pool: g[batch[n]] += x[n]
__global__ void k_pool(const float* __restrict__ x,
                       const int* __restrict__ batch,
                       float* __restrict__ g) {
  int t = blockIdx.x * blockDim.x + threadIdx.x;
  int node = t >> 5, q = t & 31;
  if (node >= NN) return;
  int b = batch[node];
  float4 v = ((const float4*)(x + (size_t)node * DD))[q];
  float* o = g + (size_t)b * DD + (size_t)q * 4;
  atomicAdd(o + 0, v.x);
  atomicAdd(o + 1, v.y);
  atomicAdd(o + 2, v.z);
  atomicAdd(o + 3, v.w);
}

// ---------------- fused GIN MLP (WMMA fp32 16x16x4) ----------------
// Block: 256 threads (8 wave32), 64 node rows per block.
// LDS buffer holds h (then h1) with padded row stride for conflict-free
// column reads of A fragments.
#define SH_STRIDE (DD + 4)

__global__ __launch_bounds__(256)
void k_gin_mlp(const float* __restrict__ x, const float* __restrict__ agg,
               const float* __restrict__ W1, const float* __restrict__ b1,
               const float* __restrict__ gm, const float* __restrict__ bt,
               const float* __restrict__ mn, const float* __restrict__ vr,
               const float* __restrict__ W2, const float* __restrict__ b2,
               float* __restrict__ xout) {
  __shared__ float sh[64][SH_STRIDE];

  const int tid = threadIdx.x;
  const int row0 = blockIdx.x * 64;

  // stage h = x + agg  (2048 float4 loads across 256 threads)
  for (int it = 0; it < 8; ++it) {
    int idx = tid + it * 256;
    int r = idx >> 5, c4 = idx & 31;
    int node = row0 + r;
    float4 v = make_float4(0.f, 0.f, 0.f, 0.f);
    if (node < NN) {
      float4 a = ((const float4*)(x + (size_t)node * DD))[c4];
      float4 b = ((const float4*)(agg + (size_t)node * DD))[c4];
      v = make_float4(a.x + b.x, a.y + b.y, a.z + b.z, a.w + b.w);
    }
    *(float4*)&sh[r][c4 * 4] = v;
  }
  __syncthreads();

  const int wave = tid >> 5;
  const int lane = tid & 31;
  const int n0 = wave * 16;       // column tile owned by this wave
  const int nn = lane & 15;       // row-within-tile (A) / column (B,C)
  const int half = lane >> 4;     // lane half selects K pair / row offset
  const int col = n0 + nn;

  // ---- GEMM1: acc = h @ W1 ----
  v8f acc0 = {}, acc1 = {}, acc2 = {}, acc3 = {};
  for (int k0 = 0; k0 < DD; k0 += 4) {
    int kb = k0 + half * 2;
    v2f b;
    b.x = W1[(size_t)(kb + 0) * DD + col];
    b.y = W1[(size_t)(kb + 1) * DD + col];
    v2f a;
    a.x = sh[0 * 16 + nn][kb]; a.y = sh[0 * 16 + nn][kb + 1];
    acc0 = __builtin_amdgcn_wmma_f32_16x16x4_f32(false, a, false, b, (short)0, acc0, false, false);
    a.x = sh[1 * 16 + nn][kb]; a.y = sh[1 * 16 + nn][kb + 1];
    acc1 = __builtin_amdgcn_wmma_f32_16x16x4_f32(false, a, false, b, (short)0, acc1, false, false);
    a.x = sh[2 * 16 + nn][kb]; a.y = sh[2 * 16 + nn][kb + 1];
    acc2 = __builtin_amdgcn_wmma_f32_16x16x4_f32(false, a, false, b, (short)0, acc2, false, false);
    a.x = sh[3 * 16 + nn][kb]; a.y = sh[3 * 16 + nn][kb + 1];
    acc3 = __builtin_amdgcn_wmma_f32_16x16x4_f32(false, a, false, b, (short)0, acc3, false, false);
  }
  __syncthreads();   // everyone done reading sh before we overwrite it

  // fused epilogue: +bias -> BatchNorm -> ReLU, write h1 back into sh
  {
    float inv = rsqrtf(vr[col] + BN_EPS) * gm[col];
    float shf = (b1[col] - mn[col]) * inv + bt[col];
#pragma unroll
    for (int r = 0; r < 8; ++r) {
      sh[0 * 16 + r + half * 8][col] = fmaxf(acc0[r] * inv + shf, 0.f);
      sh[1 * 16 + r + half * 8][col] = fmaxf(acc1[r] * inv + shf, 0.f);
      sh[2 * 16 + r + half * 8][col] = fmaxf(acc2[r] * inv + shf, 0.f);
      sh[3 * 16 + r + half * 8][col] = fmaxf(acc3[r] * inv + shf, 0.f);
    }
  }
  __syncthreads();

  // ---- GEMM2: acc = h1 @ W2 ----
  acc0 = (v8f){}; acc1 = (v8f){}; acc2 = (v8f){}; acc3 = (v8f){};
  for (int k0 = 0; k0 < DD; k0 += 4) {
    int kb = k0 + half * 2;
    v2f b;
    b.x = W2[(size_t)(kb + 0) * DD + col];
    b.y = W2[(size_t)(kb + 1) * DD + col];
    v2f a;
    a.x = sh[0 * 16 + nn][kb]; a.y = sh[0 * 16 + nn][kb + 1];
    acc0 = __builtin_amdgcn_wmma_f32_16x16x4_f32(false, a, false, b, (short)0, acc0, false, false);
    a.x = sh[1 * 16 + nn][kb]; a.y = sh[1 * 16 + nn][kb + 1];
    acc1 = __builtin_amdgcn_wmma_f32_16x16x4_f32(false, a, false, b, (short)0, acc1, false, false);
    a.x = sh[2 * 16 + nn][kb]; a.y = sh[2 * 16 + nn][kb + 1];
    acc2 = __builtin_amdgcn_wmma_f32_16x16x4_f32(false, a, false, b, (short)0, acc2, false, false);
    a.x = sh[3 * 16 + nn][kb]; a.y = sh[3 * 16 + nn][kb + 1];
    acc3 = __builtin_amdgcn_wmma_f32_16x16x4_f32(false, a, false, b, (short)0, acc3, false, false);
  }

  // epilogue: +bias -> ReLU -> store next-layer x
  {
    float bb = b2[col];
#pragma unroll
    for (int r = 0; r < 8; ++r) {
      int rr = r + half * 8;
      int node;
      node = row0 + 0 * 16 + rr;
      if (node < NN) xout[(size_t)node * DD + col] = fmaxf(acc0[r] + bb, 0.f);
      node = row0 + 1 * 16 + rr;
      if (node < NN) xout[(size_t)node * DD + col] = fmaxf(acc1[r] + bb, 0.f);
      node = row0 + 2 * 16 + rr;
      if (node < NN) xout[(size_t)node * DD + col] = fmaxf(acc2[r] + bb, 0.f);
      node = row0 + 3 * 16 + rr;
      if (node < NN) xout[(size_t)node * DD + col] = fmaxf(acc3[r] + bb, 0.f);
    }
  }
}

// ---------------- final head MLP ----------------
// grid of 4 blocks x 64 graph rows. GEMM1 via WMMA, GEMM2 (D->10) scalar.
__global__ __launch_bounds__(256)
void k_final(const float* __restrict__ g, const float* __restrict__ W1,
             const float* __restrict__ b1, const float* __restrict__ W2,
             const float* __restrict__ b2, float* __restrict__ out) {
  __shared__ float sh[64][SH_STRIDE];
  const int tid = threadIdx.x;
  const int row0 = blockIdx.x * 64;

  for (int it = 0; it < 8; ++it) {
    int idx = tid + it * 256;
    int r = idx >> 5, c4 = idx & 31;
    float4 v = ((const float4*)(g + (size_t)(row0 + r) * DD))[c4];
    *(float4*)&sh[r][c4 * 4] = v;
  }
  __syncthreads();

  const int wave = tid >> 5;
  const int lane = tid & 31;
  const int n0 = wave * 16;
  const int nn = lane & 15;
  const int half = lane >> 4;
  const int col = n0 + nn;

  v8f acc0 = {}, acc1 = {}, acc2 = {}, acc3 = {};
  for (int k0 = 0; k0 < DD; k0 += 4) {
    int kb = k0 + half * 2;
    v2f b;
    b.x = W1[(size_t)(kb + 0) * DD + col];
    b.y = W1[(size_t)(kb + 1) * DD + col];
    v2f a;
    a.x = sh[0 * 16 + nn][kb]; a.y = sh[0 * 16 + nn][kb + 1];
    acc0 = __builtin_amdgcn_wmma_f32_16x16x4_f32(false, a, false, b, (short)0, acc0, false, false);
    a.x = sh[1 * 16 + nn][kb]; a.y = sh[1 * 16 + nn][kb + 1];
    acc1 = __builtin_amdgcn_wmma_f32_16x16x4_f32(false, a, false, b, (short)0, acc1, false, false);
    a.x = sh[2 * 16 + nn][kb]; a.y = sh[2 * 16 + nn][kb + 1];
    acc2 = __builtin_amdgcn_wmma_f32_16x16x4_f32(false, a, false, b, (short)0, acc2, false, false);
    a.x = sh[3 * 16 + nn][kb]; a.y = sh[3 * 16 + nn][kb + 1];
    acc3 = __builtin_amdgcn_wmma_f32_16x16x4_f32(false, a, false, b, (short)0, acc3, false, false);
  }
  __syncthreads();

  {
    float bb = b1[col];
#pragma unroll
    for (int r = 0; r < 8; ++r) {
      sh[0 * 16 + r + half * 8][col] = fmaxf(acc0[r] + bb, 0.f);
      sh[1 * 16 + r + half * 8][col] = fmaxf(acc1[r] + bb, 0.f);
      sh[2 * 16 + r + half * 8][col] = fmaxf(acc2[r] + bb, 0.f);
      sh[3 * 16 + r + half * 8][col] = fmaxf(acc3[r] + bb, 0.f);
    }
  }
  __syncthreads();

  // GEMM2: 64 rows x 10 classes, K=128 scalar dots from LDS
  for (int o = tid; o < 64 * CC; o += 256) {
    int r = o / CC, c = o % CC;
    float s = b2[c];
    for (int k = 0; k < DD; ++k) s += sh[r][k] * W2[(size_t)k * CC + c];
    out[(size_t)(row0 + r) * CC + c] = s;
  }
}

// ---------------- host-side orchestration ----------------

extern "C" void kernel_launch(void* const* d_in, const int* in_sizes, int n_in,
                              void* d_out, int out_size, void* d_ws, size_t ws_size,
                              hipStream_t stream) {
  const float* x     = (const float*)d_in[0];
  const int*   ei    = (const int*)d_in[1];   // [2,E]: src then dst
  const int*   smask = (const int*)d_in[2];
  const float* emask = (const float*)d_in[3];
  const int*   batch = (const int*)d_in[4];
  const float* Ws1   = (const float*)d_in[5];
  const float* bs1   = (const float*)d_in[6];
  const float* gamma = (const float*)d_in[7];
  const float* beta  = (const float*)d_in[8];
  const float* mean  = (const float*)d_in[9];
  const float* var   = (const float*)d_in[10];
  const float* Ws2   = (const float*)d_in[11];
  const float* bs2   = (const float*)d_in[12];
  const float* Wf1   = (const float*)d_in[13];
  const float* bf1   = (const float*)d_in[14];
  const float* Wf2   = (const float*)d_in[15];
  const float* bf2   = (const float*)d_in[16];
  float* out = (float*)d_out;

  float* bufA = (float*)d_ws;                       // current node features
  float* bufB = bufA + (size_t)NN * DD;             // scatter accumulator
  float* gbuf = bufB + (size_t)NN * DD;             // pooled [G,D]

  const int* src = ei;
  const int* dst = ei + EE;

  dim3 blk(256);
  const int nzero = NN * DD;
  const int zgrid = (nzero + 255) / 256;
  const int sgrid = (int)(((size_t)EE * 32 + 255) / 256);
  const int ngrid = (int)(((size_t)NN * 32 + 255) / 256);
  const int mgrid = (NN + 63) / 64;

  // SimpleConv('add') with edge weights, then supernode select
  k_zero<<<zgrid, blk, 0, stream>>>(bufB, nzero);
  k_scatter<<<sgrid, blk, 0, stream>>>(x, src, dst, emask, bufB);
  k_select<<<ngrid, blk, 0, stream>>>(x, bufB, smask, bufA);

  // 3 GIN layers: aggregate then fused MLP (in-place on bufA)
  for (int i = 0; i < LL; ++i) {
    k_zero<<<zgrid, blk, 0, stream>>>(bufB, nzero);
    k_scatter<<<sgrid, blk, 0, stream>>>(bufA, src, dst, (const float*)nullptr, bufB);
    k_gin_mlp<<<mgrid, blk, 0, stream>>>(
        bufA, bufB,
        Ws1 + (size_t)i * DD * DD, bs1 + (size_t)i * DD,
        gamma + (size_t)i * DD, beta + (size_t)i * DD,
        mean + (size_t)i * DD, var + (size_t)i * DD,
        Ws2 + (size_t)i * DD * DD, bs2 + (size_t)i * DD,
        bufA);
  }

  // global add pool + head MLP
  k_zero<<<(GG * DD + 255) / 256, blk, 0, stream>>>(gbuf, GG * DD);
  k_pool<<<ngrid, blk, 0, stream>>>(bufA, batch, gbuf);
  k_final<<<GG / 64, blk, 0, stream>>>(gbuf, Wf1, bf1, Wf2, bf2, out);
}